// MaxNormPooling2D_12524124635604
// MI455X (gfx1250) — compile-verified
//
#include <hip/hip_runtime.h>

// Quaternion max-norm 2x2 pooling, B=16 H=128 W=128 C=64 (NHWC), pool=stride=2.
// Pure bandwidth kernel: 256MB read + 64MB write -> ~13.7us floor @ 23.3TB/s.
// Strategy: argmax over squared norms (sqrt dropped; monotone, tie-preserving),
// b128 coalesced traffic with non-temporal hints (zero reuse, working set >
// 192MB L2), and the CDNA5 async global->LDS path (ASYNCcnt) with a partial
// s_wait_asynccnt pipeline: compute on the first 8 staged tiles while the last
// 8 async transfers are still in flight.

typedef float f4  __attribute__((ext_vector_type(4)));
typedef int   v4i __attribute__((ext_vector_type(4)));

#define OUT_COMP_STRIDE 4194304u   // 16*64*64*64 floats per output component
#define NTHREADS_TOTAL  1048576u   // one thread per float4 "site" (4 channels of one out pixel)

#if __has_builtin(__builtin_amdgcn_global_load_async_to_lds_b128)
#define HAVE_ASYNC_LDS 1
#else
#define HAVE_ASYNC_LDS 0
#endif

#if __has_builtin(__builtin_amdgcn_s_wait_asynccnt)
#define WAIT_ASYNC(n) __builtin_amdgcn_s_wait_asynccnt(n)
#else
#define WAIT_ASYNC(n) asm volatile("s_wait_asynccnt %0" ::"i"(n) : "memory")
#endif

// t -> base element offset of window position (0,0) for this site; also returns pixel id.
__device__ __forceinline__ unsigned site_base(unsigned t, unsigned& pix) {
  unsigned c4  = t & 15u;          // which float4 group of the 64 channels
  pix          = t >> 4;           // b*4096 + ho*64 + wo
  unsigned wo  = pix & 63u;
  unsigned hob = pix >> 6;         // b*64 + ho
  unsigned ho  = hob & 63u;
  unsigned b   = hob >> 6;
  unsigned h = ho * 2u, w = wo * 2u;
  return ((b * 128u + h) * 128u + w) * 64u + c4 * 4u;
}

// Window-position element offsets in row-major (dh, dw) order: (0,0),(0,1),(1,0),(1,1).
// Matches tf/jnp argmax flatten order for correct first-max tie-break.
__device__ __forceinline__ unsigned win_off(int pos) {
  const unsigned off[4] = {0u, 64u, 8192u, 8256u};   // dw:+64, dh:+128*64
  return off[pos];
}

__device__ __forceinline__ void select_pos(int pos, const f4 a[4], f4& bn, f4 best[4]) {
  f4 n = a[0] * a[0] + a[1] * a[1] + a[2] * a[2] + a[3] * a[3];
  if (pos == 0) {
    bn = n;
    best[0] = a[0]; best[1] = a[1]; best[2] = a[2]; best[3] = a[3];
  } else {
#pragma unroll
    for (int j = 0; j < 4; ++j) {
      bool gt = n[j] > bn[j];                 // strict > keeps first max (tie-break)
      bn[j]      = gt ? n[j]    : bn[j];
      best[0][j] = gt ? a[0][j] : best[0][j];
      best[1][j] = gt ? a[1][j] : best[1][j];
      best[2][j] = gt ? a[2][j] : best[2][j];
      best[3][j] = gt ? a[3][j] : best[3][j];
    }
  }
}

__device__ __forceinline__ void store_out(float* __restrict__ out, unsigned t,
                                          unsigned pix, const f4 best[4]) {
  unsigned o = pix * 64u + (t & 15u) * 4u;
#pragma unroll
  for (int c = 0; c < 4; ++c)
    __builtin_nontemporal_store(best[c], (f4*)(out + (size_t)c * OUT_COMP_STRIDE + o));
}

#if HAVE_ASYNC_LDS
// ---------------- CDNA5 path: async global->LDS (ASYNCcnt pipeline) ----------------
__global__ __launch_bounds__(256) void pool_async(
    const float* __restrict__ x0, const float* __restrict__ x1,
    const float* __restrict__ x2, const float* __restrict__ x3,
    float* __restrict__ out) {
  // [slot][tid] layout: a wave's ds_load_b128 of one slot walks banks linearly.
  __shared__ f4 smem[16][256];                       // 64 KB

  unsigned tid = threadIdx.x;
  unsigned t   = blockIdx.x * 256u + tid;
  unsigned pix;
  unsigned base = site_base(t, pix);
  const float* xs[4] = {x0, x1, x2, x3};

  // Issue all 16 async b128 loads per thread, pos-major; no destination VGPRs
  // held while in flight. cpol=1 (TH_NT): stream-once input, don't cache in L2.
#pragma unroll
  for (int pos = 0; pos < 4; ++pos) {
#pragma unroll
    for (int c = 0; c < 4; ++c) {
      const float* g = xs[c] + base + win_off(pos);
      __builtin_amdgcn_global_load_async_to_lds_b128(
          (__attribute__((address_space(1))) v4i*)(v4i*)g,
          (__attribute__((address_space(3))) v4i*)(v4i*)&smem[pos * 4 + c][tid],
          /*offset=*/0, /*cpol=*/1 /*TH_NT*/);
    }
  }

  f4 bn, best[4];

  // Async loads retire in issue order -> asynccnt<=8 guarantees slots 0..7
  // (pos 0 and 1) are resident in LDS while pos 2/3 transfers continue.
  WAIT_ASYNC(8);
  asm volatile("" ::: "memory");
#pragma unroll
  for (int pos = 0; pos < 2; ++pos) {
    f4 a[4];
#pragma unroll
    for (int c = 0; c < 4; ++c)
      a[c] = smem[pos * 4 + c][tid];
    select_pos(pos, a, bn, best);
  }

  WAIT_ASYNC(0);
  asm volatile("" ::: "memory");
#pragma unroll
  for (int pos = 2; pos < 4; ++pos) {
    f4 a[4];
#pragma unroll
    for (int c = 0; c < 4; ++c)
      a[c] = smem[pos * 4 + c][tid];
    select_pos(pos, a, bn, best);
  }

  store_out(out, t, pix, best);
}

#else  // !HAVE_ASYNC_LDS
// ---------------- Fallback: direct NT b128 loads ----------------
__global__ __launch_bounds__(256) void pool_direct(
    const float* __restrict__ x0, const float* __restrict__ x1,
    const float* __restrict__ x2, const float* __restrict__ x3,
    float* __restrict__ out) {
  unsigned t = blockIdx.x * 256u + threadIdx.x;
  unsigned pix;
  unsigned base = site_base(t, pix);
  const float* xs[4] = {x0, x1, x2, x3};

  f4 bn, best[4];
#pragma unroll
  for (int pos = 0; pos < 4; ++pos) {
    f4 a[4];
#pragma unroll
    for (int c = 0; c < 4; ++c)
      a[c] = __builtin_nontemporal_load((const f4*)(xs[c] + base + win_off(pos)));
    select_pos(pos, a, bn, best);
  }
  store_out(out, t, pix, best);
}
#endif  // HAVE_ASYNC_LDS

extern "C" void kernel_launch(void* const* d_in, const int* /*in_sizes*/, int /*n_in*/,
                              void* d_out, int /*out_size*/, void* /*d_ws*/, size_t /*ws_size*/,
                              hipStream_t stream) {
  const float* x0 = (const float*)d_in[0];
  const float* x1 = (const float*)d_in[1];
  const float* x2 = (const float*)d_in[2];
  const float* x3 = (const float*)d_in[3];
  float* out = (float*)d_out;

  dim3 block(256);
  dim3 grid(NTHREADS_TOTAL / 256u);   // 4096 blocks
#if HAVE_ASYNC_LDS
  pool_async<<<grid, block, 0, stream>>>(x0, x1, x2, x3, out);
#else
  pool_direct<<<grid, block, 0, stream>>>(x0, x1, x2, x3, out);
#endif
}